// RNNNode_8959301780015
// MI455X (gfx1250) — compile-verified
//
#include <hip/hip_runtime.h>

// ---------------------------------------------------------------------------
// Diagonal linear RNN:  h_t = h_{t-1} * a + x_t @ b ;  y_t = h_t @ c
// T=4096, B=64, D_IN=64, D_HID=512, D_OUT=64, all fp32.
//
// 3-phase chunked parallel scan, fp32 WMMA (v_wmma_f32_16x16x4_f32).
// 128 chunks x 32 steps. Kernel A is register-only (no LDS); kernel C keeps h
// in a 64x516-padded LDS tile for the K=512 output GEMM. b/c weights are
// pre-paired into float2 fragments so every WMMA B-operand is one b64 load.
// An opaque SGPR zero bumps the weight pointers each step so LICM cannot
// hoist (and then spill) the loop-invariant fragment loads.
// ---------------------------------------------------------------------------

typedef __attribute__((ext_vector_type(2))) float v2f;
typedef __attribute__((ext_vector_type(8))) float v8f;

#define T_STEPS 4096
#define BATCH   64
#define DIN     64
#define DHID    512
#define DOUT    64
#define CHUNKS  128
#define CLEN    32            // T_STEPS / CHUNKS
#define HSTRIDE 516           // padded LDS row stride (floats); 516 % 64 = 4 -> conflict-free
#define NWAVES  16
#define NTHREADS (NWAVES * 32)

__device__ __forceinline__ v8f wmma4(v2f a, v2f b, v8f c) {
    return __builtin_amdgcn_wmma_f32_16x16x4_f32(
        false, a, false, b, (short)0, c, false, false);
}

// Opaque zero the compiler cannot constant-fold: used to bump pointers inside
// loops so loop-invariant fragment loads are re-issued instead of hoisted+spilled.
__device__ __forceinline__ int opaque_zero() {
    int z;
    asm volatile("s_mov_b32 %0, 0" : "=s"(z));
    return z;
}

// -------------------------------------------------------------------------
// Kernel P: pre-pair weights for WMMA B-fragments.
//   bp[p*DHID + n] = { b[2p][n], b[2p+1][n] }   p in [0,32)
//   cp[p*DOUT + n] = { c[2p][n], c[2p+1][n] }   p in [0,256)
// -------------------------------------------------------------------------
__global__ __launch_bounds__(256)
void rnn_pack(const float* __restrict__ bW, const float* __restrict__ cW,
              v2f* __restrict__ bp, v2f* __restrict__ cp)
{
    const int i = blockIdx.x * 256 + threadIdx.x;      // 0 .. 16383
    if (i < 32 * DHID) {
        const int p = i >> 9, n = i & (DHID - 1);
        v2f t;
        t.x = bW[(size_t)(2 * p) * DHID + n];
        t.y = bW[(size_t)(2 * p + 1) * DHID + n];
        bp[i] = t;
    }
    if (i < 256 * DOUT) {
        const int p = i >> 6, n = i & (DOUT - 1);
        v2f t;
        t.x = cW[(size_t)(2 * p) * DOUT + n];
        t.y = cW[(size_t)(2 * p + 1) * DOUT + n];
        cp[i] = t;
    }
}

// -------------------------------------------------------------------------
// Kernel A: per-chunk decayed summary  S_j[b,h] = sum_i a^(CLEN-1-i) * xb_i.
// Entirely in registers: Sacc lives in C/D fragment layout, one n-tile pair
// of transient WMMA accumulators at a time. No LDS, no barriers.
// -------------------------------------------------------------------------
__global__ __launch_bounds__(NTHREADS)
void rnn_chunk_summary(const float* __restrict__ x,
                       const float* __restrict__ aVec,
                       const v2f* __restrict__ bp0,
                       float* __restrict__ S)
{
    const int chunk = blockIdx.x;
    const int tid   = threadIdx.x;
    const int wave  = tid >> 5;
    const int lane  = tid & 31;
    const int lmod  = lane & 15;
    const int lhi   = lane >> 4;

    const int m_tile = wave & 3;            // 4 x 16 batch rows
    const int nclus  = wave >> 2;           // 4 clusters x 8 n-tiles (512 cols)
    const int rowA   = m_tile * 16 + lmod;

    float av[8];
    #pragma unroll
    for (int nt = 0; nt < 8; ++nt)
        av[nt] = aVec[(nclus * 8 + nt) * 16 + lmod];

    const int zero = opaque_zero();
    const v2f* bp = bp0;

    v8f Sacc[8] = {};

    for (int i = 0; i < CLEN; ++i) {
        bp += zero;                         // defeat LICM: re-load b frags each step
        const float* xt = x + (size_t)(chunk * CLEN + i) * BATCH * DIN;

        v2f afrag[16];
        #pragma unroll
        for (int kt = 0; kt < 16; ++kt)
            afrag[kt] = *(const v2f*)(xt + rowA * DIN + kt * 4 + lhi * 2);

        #pragma unroll
        for (int ntp = 0; ntp < 4; ++ntp) {
            const int n0 = (nclus * 8 + ntp * 2) * 16 + lmod;
            v8f acc0 = {}, acc1 = {};
            #pragma unroll
            for (int kt = 0; kt < 16; ++kt) {
                const int p = kt * 2 + lhi;
                const v2f b0 = bp[p * DHID + n0];
                const v2f b1 = bp[p * DHID + n0 + 16];
                acc0 = wmma4(afrag[kt], b0, acc0);
                acc1 = wmma4(afrag[kt], b1, acc1);
            }
            const float a0 = av[ntp * 2], a1 = av[ntp * 2 + 1];
            #pragma unroll
            for (int r = 0; r < 8; ++r) {
                Sacc[ntp * 2][r]     = Sacc[ntp * 2][r]     * a0 + acc0[r];
                Sacc[ntp * 2 + 1][r] = Sacc[ntp * 2 + 1][r] * a1 + acc1[r];
            }
        }
    }

    float* Sj = S + (size_t)chunk * BATCH * DHID;
    #pragma unroll
    for (int nt = 0; nt < 8; ++nt) {
        const int col = (nclus * 8 + nt) * 16 + lmod;
        #pragma unroll
        for (int r = 0; r < 8; ++r) {
            const int row = m_tile * 16 + r + lhi * 8;
            Sj[row * DHID + col] = Sacc[nt][r];
        }
    }
}

// -------------------------------------------------------------------------
// Kernel B: sequential carry scan over the 128 chunk summaries.
// C_0 = h0 ; C_{j+1} = a^CLEN * C_j + S_j ; h_final = C_CHUNKS.
// -------------------------------------------------------------------------
__global__ __launch_bounds__(256)
void rnn_carry_scan(const float* __restrict__ h0,
                    const float* __restrict__ aVec,
                    const float* __restrict__ S,
                    float* __restrict__ Cout,
                    float* __restrict__ hfinal)
{
    const int idx = blockIdx.x * blockDim.x + threadIdx.x;
    if (idx >= BATCH * DHID) return;
    const int hcol = idx & (DHID - 1);

    float aL = aVec[hcol];
    #pragma unroll
    for (int s = 0; s < 5; ++s) aL *= aL;   // a^32 (CLEN = 32)

    float c = h0[idx];
    for (int j = 0; j < CHUNKS; ++j) {
        Cout[(size_t)j * BATCH * DHID + idx] = c;
        c = aL * c + S[(size_t)j * BATCH * DHID + idx];
    }
    hfinal[idx] = c;
}

// -------------------------------------------------------------------------
// Kernel C: per chunk, recompute GEMM1, exact scan from carry C_j in LDS,
// then y_t = h_t @ c with two interleaved WMMA chains (K split in halves).
// -------------------------------------------------------------------------
__global__ __launch_bounds__(NTHREADS)
void rnn_chunk_emit(const float* __restrict__ x,
                    const float* __restrict__ aVec,
                    const v2f* __restrict__ bp0,
                    const v2f* __restrict__ cp0,
                    const float* __restrict__ Cin,
                    float* __restrict__ y)
{
    extern __shared__ float hbuf[];         // 64 x HSTRIDE fp32

    const int chunk = blockIdx.x;
    const int tid   = threadIdx.x;
    const int wave  = tid >> 5;
    const int lane  = tid & 31;
    const int lmod  = lane & 15;
    const int lhi   = lane >> 4;

    const int m_tile = wave & 3;
    const int nclus  = wave >> 2;
    const int rowA   = m_tile * 16 + lmod;
    const int n2     = wave >> 2;           // GEMM2: 4 x 4 output tiles
    const int rowH   = m_tile * 16 + lmod;
    const int ncol   = n2 * 16 + lmod;

    float av[8];
    #pragma unroll
    for (int nt = 0; nt < 8; ++nt)
        av[nt] = aVec[(nclus * 8 + nt) * 16 + lmod];

    // h := carry for this chunk
    const float* Cj = Cin + (size_t)chunk * BATCH * DHID;
    for (int idx = tid; idx < BATCH * DHID; idx += NTHREADS) {
        const int row = idx >> 9;
        const int col = idx & (DHID - 1);
        hbuf[row * HSTRIDE + col] = Cj[idx];
    }
    __syncthreads();

    const int zero = opaque_zero();
    const v2f* bp = bp0;
    const v2f* cp = cp0;

    for (int i = 0; i < CLEN; ++i) {
        bp += zero;                         // defeat LICM on weight-fragment loads
        cp += zero;
        const int t = chunk * CLEN + i;
        const float* xt = x + (size_t)t * BATCH * DIN;

        v2f afrag[16];
        #pragma unroll
        for (int kt = 0; kt < 16; ++kt)
            afrag[kt] = *(const v2f*)(xt + rowA * DIN + kt * 4 + lhi * 2);

        __syncthreads();    // previous step's GEMM2 reads of hbuf are done

        // ---- GEMM1 + scan, one n-tile pair at a time ----
        #pragma unroll
        for (int ntp = 0; ntp < 4; ++ntp) {
            const int n0 = (nclus * 8 + ntp * 2) * 16 + lmod;
            v8f acc0 = {}, acc1 = {};
            #pragma unroll
            for (int kt = 0; kt < 16; ++kt) {
                const int p = kt * 2 + lhi;
                const v2f b0 = bp[p * DHID + n0];
                const v2f b1 = bp[p * DHID + n0 + 16];
                acc0 = wmma4(afrag[kt], b0, acc0);
                acc1 = wmma4(afrag[kt], b1, acc1);
            }
            const float a0 = av[ntp * 2], a1 = av[ntp * 2 + 1];
            #pragma unroll
            for (int r = 0; r < 8; ++r) {
                const int row = m_tile * 16 + r + lhi * 8;
                const int i0  = row * HSTRIDE + n0;
                hbuf[i0]      = hbuf[i0]      * a0 + acc0[r];
                hbuf[i0 + 16] = hbuf[i0 + 16] * a1 + acc1[r];
            }
        }
        __syncthreads();    // h_t visible to all waves

        // ---- GEMM2: y_t = h_t @ c, two chains (K 0..255 | 256..511) ----
        v8f y0 = {}, y1 = {};
        #pragma unroll 8
        for (int kt = 0; kt < 64; ++kt) {
            const int kbA = kt * 4 + lhi * 2;
            const int kbB = kbA + 256;
            const v2f hA = *(const v2f*)(hbuf + rowH * HSTRIDE + kbA);
            const v2f hB = *(const v2f*)(hbuf + rowH * HSTRIDE + kbB);
            const v2f cA = cp[(kbA >> 1) * DOUT + ncol];
            const v2f cB = cp[(kbB >> 1) * DOUT + ncol];
            y0 = wmma4(hA, cA, y0);
            y1 = wmma4(hB, cB, y1);
        }

        float* yt = y + (size_t)t * BATCH * DOUT;
        #pragma unroll
        for (int r = 0; r < 8; ++r) {
            const int row = m_tile * 16 + r + lhi * 8;
            yt[row * DOUT + ncol] = y0[r] + y1[r];
        }
    }
}

// -------------------------------------------------------------------------
// Host launcher. Inputs (setup_inputs order): h[64,512], x[4096,64,64],
// a[512], b[64,512], c[512,64]. Output: h_final (32768 floats) ++ y
// (4096*64*64 floats). Workspace: S (16MB) + carries (16MB) + paired b/c.
// -------------------------------------------------------------------------
extern "C" void kernel_launch(void* const* d_in, const int* in_sizes, int n_in,
                              void* d_out, int out_size, void* d_ws, size_t ws_size,
                              hipStream_t stream)
{
    const float* h0 = (const float*)d_in[0];
    const float* x  = (const float*)d_in[1];
    const float* a  = (const float*)d_in[2];
    const float* bW = (const float*)d_in[3];
    const float* cW = (const float*)d_in[4];

    float* out    = (float*)d_out;
    float* hfinal = out;                        // [64, 512]
    float* y      = out + BATCH * DHID;         // [4096, 64, 64]

    float* S  = (float*)d_ws;                            // CHUNKS*64*512
    float* C  = S + (size_t)CHUNKS * BATCH * DHID;       // CHUNKS*64*512
    v2f*   bp = (v2f*)(C + (size_t)CHUNKS * BATCH * DHID);   // 32*512 pairs
    v2f*   cp = bp + 32 * DHID;                              // 256*64 pairs

    const size_t shmem = (size_t)(64 * HSTRIDE) * sizeof(float);

    rnn_pack<<<64, 256, 0, stream>>>(bW, cW, bp, cp);
    rnn_chunk_summary<<<CHUNKS, NTHREADS, 0, stream>>>(x, a, bp, S);
    rnn_carry_scan<<<(BATCH * DHID + 255) / 256, 256, 0, stream>>>(h0, a, S, C, hfinal);
    rnn_chunk_emit<<<CHUNKS, NTHREADS, shmem, stream>>>(x, a, bp, cp, C, y);
}